// MLSTMBlock_69114613730533
// MI455X (gfx1250) — compile-verified
//
#include <hip/hip_runtime.h>
#include <hip/hip_bf16.h>
#include <math.h>

// ---------------------------------------------------------------------------
// mLSTM block, single step.  B=64, INPUT_DIM=1024, HIDDEN=2048, HEADS=4,
// PROJ=2048, HEAD_DIM=512.
// Roofline: C state read+write = 537MB  (~23us @ 23.3TB/s)  -> c_update_kernel
//           GEMM weights       = ~108MB (~4.6us)            -> WMMA f32 GEMMs
// GEMM staging uses GLOBAL_LOAD_ASYNC_TO_LDS_B128 (ASYNCcnt) with LDS
// double-buffering when available; float4 reg-staging otherwise.
// ---------------------------------------------------------------------------

typedef float v2f __attribute__((ext_vector_type(2)));
typedef float v8f __attribute__((ext_vector_type(8)));
typedef int   v4i_vs __attribute__((vector_size(16)));          // int4 as in builtin sig
typedef __attribute__((address_space(1))) v4i_vs* glob128_t;    // global int4*
typedef __attribute__((address_space(3))) v4i_vs* lds128_t;     // LDS int4*

namespace {
constexpr int BATCH   = 64;
constexpr int IN_DIM  = 1024;
constexpr int HIDDEN  = 2048;
constexpr int HEADS   = 4;
constexpr int PROJ    = 2048;
constexpr int HD      = 512;              // head dim
constexpr int UP_ROWS = PROJ + HIDDEN;    // 4096
constexpr float EPSV  = 1e-5f;

constexpr int KCHUNK = 64;                // K staged per iteration
constexpr int APAD   = 68;                // LDS row stride: 68%64==4 -> 16 rows
                                          // hit 16 distinct banks; 272B keeps
                                          // float4 alignment
}

#if __has_builtin(__builtin_amdgcn_global_load_async_to_lds_b128)
#define USE_ASYNC_LDS 1
#else
#define USE_ASYNC_LDS 0
#endif

__device__ __forceinline__ void wait_asynccnt0() {
#if __has_builtin(__builtin_amdgcn_s_wait_asynccnt)
    __builtin_amdgcn_s_wait_asynccnt(0);
#else
    asm volatile("s_wait_asynccnt 0x0" ::: "memory");
#endif
}

#if USE_ASYNC_LDS
__device__ __forceinline__ void async_copy_b128(const float* g, float* l) {
    __builtin_amdgcn_global_load_async_to_lds_b128(
        (glob128_t)(uintptr_t)g,
        (lds128_t)(uintptr_t)l, 0, 0);
}
#endif

// ---------------- block-wide sum reduction (blockDim.x == 256) -------------
__device__ __forceinline__ float block_reduce_sum(float v, float* red) {
    const int tid = threadIdx.x;
    red[tid] = v;
    __syncthreads();
    for (int off = 128; off > 0; off >>= 1) {
        if (tid < off) red[tid] += red[tid + off];
        __syncthreads();
    }
    float r = red[0];
    __syncthreads();
    return r;
}

// ---------------- 1. LayerNorm of x  [64,1024] -----------------------------
__global__ __launch_bounds__(256) void layernorm_kernel(
    const float* __restrict__ x, const float* __restrict__ w,
    const float* __restrict__ b, float* __restrict__ xn)
{
    __shared__ float red[256];
    const int row = blockIdx.x;                 // 0..63
    const float* xr = x + (size_t)row * IN_DIM;

    float s = 0.f;
    for (int i = threadIdx.x; i < IN_DIM; i += 256) s += xr[i];
    const float mu = block_reduce_sum(s, red) * (1.0f / IN_DIM);

    float v = 0.f;
    for (int i = threadIdx.x; i < IN_DIM; i += 256) {
        float d = xr[i] - mu; v += d * d;
    }
    const float rstd = rsqrtf(block_reduce_sum(v, red) * (1.0f / IN_DIM) + EPSV);

    for (int i = threadIdx.x; i < IN_DIM; i += 256)
        xn[(size_t)row * IN_DIM + i] = (xr[i] - mu) * rstd * w[i] + b[i];
}

// ---------------- 2. WMMA GEMM:  out[64,N] = (A[64,K] @ W[N,K]^T + bias)*scale (+resid)
// Wave = one 16x16 tile; block of 8 waves covers M=64 x N=32.  K staged
// through LDS in 64-wide chunks; async copies + double buffer when available.
__global__ __launch_bounds__(256) void gemm_wmma_kernel(
    const float* __restrict__ A, int lda,      // [64,K], row stride lda
    const float* __restrict__ W,               // [N,K] row-major
    const float* __restrict__ bias,            // [N] or nullptr
    const float* __restrict__ resid,           // [64,N] or nullptr (stride N)
    float* __restrict__ out,                   // [64,N]
    int K, int N, float scale)
{
#if USE_ASYNC_LDS
    constexpr int NBUF = 2;
#else
    constexpr int NBUF = 1;
#endif
    __shared__ float sA[NBUF][64][APAD];
    __shared__ float sW[NBUF][32][APAD];

    const int tid  = threadIdx.x;
    const int wave = tid >> 5;                 // 0..7
    const int lane = tid & 31;
    const int half = lane >> 4;                // 0/1
    const int idx  = lane & 15;                // 0..15

    const int n0 = blockIdx.x * 32;            // 32-wide N tile per block
    const int mt = wave >> 1;                  // 0..3  -> rows mt*16
    const int nt = wave & 1;                   // 0..1  -> cols n0 + nt*16

    v8f acc = {};
    const int nchunk = K / KCHUNK;

#if USE_ASYNC_LDS
    // Per-chunk staging: A = 64 rows x 16 float4 (1024 xfers, 4/thread),
    //                    W = 32 rows x 16 float4 ( 512 xfers, 2/thread).
    auto issue = [&](int c, int buf) {
        const int k0 = c * KCHUNK;
        #pragma unroll
        for (int i = 0; i < 4; ++i) {
            const int e = tid + i * 256, r = e >> 4, c4 = e & 15;
            async_copy_b128(A + (size_t)r * lda + k0 + c4 * 4,
                            &sA[buf][r][c4 * 4]);
        }
        #pragma unroll
        for (int i = 0; i < 2; ++i) {
            const int e = tid + i * 256, r = e >> 4, c4 = e & 15;
            async_copy_b128(W + (size_t)(n0 + r) * K + k0 + c4 * 4,
                            &sW[buf][r][c4 * 4]);
        }
    };

    issue(0, 0);
    for (int c = 0; c < nchunk; ++c) {
        wait_asynccnt0();                      // my copies for chunk c done
        __syncthreads();                       // everyone's copies visible;
                                               // everyone done with other buf
        if (c + 1 < nchunk) issue(c + 1, (c + 1) & 1);
        const int buf = c & 1;
        #pragma unroll
        for (int kk = 0; kk < KCHUNK; kk += 4) {
            v2f a, b;
            a.x = sA[buf][mt * 16 + idx][kk + 2 * half];
            a.y = sA[buf][mt * 16 + idx][kk + 2 * half + 1];
            b.x = sW[buf][nt * 16 + idx][kk + 2 * half];
            b.y = sW[buf][nt * 16 + idx][kk + 2 * half + 1];
            acc = __builtin_amdgcn_wmma_f32_16x16x4_f32(
                false, a, false, b, (short)0, acc, false, false);
        }
    }
#else
    for (int c = 0; c < nchunk; ++c) {
        const int k0 = c * KCHUNK;
        float4 ra[4], rw[2];
        #pragma unroll
        for (int i = 0; i < 4; ++i) {
            const int e = tid + i * 256, r = e >> 4, c4 = e & 15;
            ra[i] = *(const float4*)(A + (size_t)r * lda + k0 + c4 * 4);
        }
        #pragma unroll
        for (int i = 0; i < 2; ++i) {
            const int e = tid + i * 256, r = e >> 4, c4 = e & 15;
            rw[i] = *(const float4*)(W + (size_t)(n0 + r) * K + k0 + c4 * 4);
        }
        __syncthreads();
        #pragma unroll
        for (int i = 0; i < 4; ++i) {
            const int e = tid + i * 256, r = e >> 4, c4 = e & 15;
            *(float4*)&sA[0][r][c4 * 4] = ra[i];
        }
        #pragma unroll
        for (int i = 0; i < 2; ++i) {
            const int e = tid + i * 256, r = e >> 4, c4 = e & 15;
            *(float4*)&sW[0][r][c4 * 4] = rw[i];
        }
        __syncthreads();
        #pragma unroll
        for (int kk = 0; kk < KCHUNK; kk += 4) {
            v2f a, b;
            a.x = sA[0][mt * 16 + idx][kk + 2 * half];
            a.y = sA[0][mt * 16 + idx][kk + 2 * half + 1];
            b.x = sW[0][nt * 16 + idx][kk + 2 * half];
            b.y = sW[0][nt * 16 + idx][kk + 2 * half + 1];
            acc = __builtin_amdgcn_wmma_f32_16x16x4_f32(
                false, a, false, b, (short)0, acc, false, false);
        }
        __syncthreads();
    }
#endif

    // Epilogue: D VGPR r -> (M = mt*16 + r + 8*half, N = n0 + nt*16 + idx)
    const int col  = n0 + nt * 16 + idx;
    const float bv = bias ? bias[col] : 0.0f;
    #pragma unroll
    for (int r = 0; r < 8; ++r) {
        const int row = mt * 16 + r + 8 * half;
        float v = (acc[r] + bv) * scale;
        if (resid) v += resid[(size_t)row * N + col];
        out[(size_t)row * N + col] = v;
    }
}

// ---------------- 3. i/f gate GEMVs: x_qkv[64,2048] . w[4,2048] ------------
__global__ __launch_bounds__(256) void if_gemv_kernel(
    const float* __restrict__ xup,             // [64,4096], qkv = cols [0,2048)
    const float* __restrict__ wi, const float* __restrict__ bi,
    const float* __restrict__ wf, const float* __restrict__ bf,
    float* __restrict__ i_tilda, float* __restrict__ f_tilda)  // each [64*4]
{
    __shared__ float red[256];
    const int o     = blockIdx.x;              // 0..511
    const int which = o >> 8;                  // 0 = i, 1 = f
    const int bh    = o & 255;
    const int b     = bh >> 2;
    const int h     = bh & 3;

    const float* w  = (which ? wf : wi) + (size_t)h * PROJ;
    const float* xr = xup + (size_t)b * UP_ROWS;

    float s = 0.f;
    for (int k = threadIdx.x; k < PROJ; k += 256) s += xr[k] * w[k];
    const float tot = block_reduce_sum(s, red);
    if (threadIdx.x == 0)
        (which ? f_tilda : i_tilda)[bh] = tot + (which ? bf : bi)[h];
}

// ---------------- 4. gates: mt / it / ft  (256 values) ---------------------
__global__ __launch_bounds__(256) void gates_kernel(
    const float* __restrict__ i_tilda, const float* __restrict__ f_tilda,
    const float* __restrict__ m_prev,
    float* __restrict__ mt_out, float* __restrict__ it_g, float* __restrict__ ft_g)
{
    const int bh = threadIdx.x;                // 0..255
    const float iv = i_tilda[bh];
    const float fv = f_tilda[bh] + m_prev[bh];
    const float m  = fmaxf(fv, iv);
    mt_out[bh] = m;
    it_g[bh]   = __expf(iv - m);
    ft_g[bh]   = __expf(fv - m);
}

// ---------------- 5. nt update + denominator -------------------------------
__global__ __launch_bounds__(256) void n_kernel(
    const float* __restrict__ n_prev,          // [64,4,512]
    const float* __restrict__ kvec,            // [64,2048] (already * DIM_NORM)
    const float* __restrict__ qvec,            // [64,2048]
    const float* __restrict__ it_g, const float* __restrict__ ft_g,
    float* __restrict__ nt_out, float* __restrict__ denom)
{
    __shared__ float red[256];
    const int bh = blockIdx.x;                 // 0..255
    const int b  = bh >> 2, h = bh & 3;
    const float it = it_g[bh], ft = ft_g[bh];
    const int base = b * HIDDEN + h * HD;

    float acc = 0.f;
    for (int j = threadIdx.x; j < HD; j += 256) {
        const float n = ft * n_prev[(size_t)bh * HD + j] + it * kvec[base + j];
        nt_out[(size_t)bh * HD + j] = n;
        acc += n * qvec[base + j];
    }
    const float tot = block_reduce_sum(acc, red);
    if (threadIdx.x == 0) denom[bh] = fmaxf(fabsf(tot), 1.0f);
}

// ---------------- 6. C state update fused with h = Ct.q  (HBM-bound) -------
// grid: (b,h,rowgroup) = 64*4*8 blocks; wave handles 8 rows of the 512x512.
__global__ __launch_bounds__(256) void c_update_kernel(
    const float* __restrict__ C_prev,
    const float* __restrict__ kvec, const float* __restrict__ qvec,
    const float* __restrict__ vvec,
    const float* __restrict__ it_g, const float* __restrict__ ft_g,
    const float* __restrict__ denom,
    float* __restrict__ Ct_out, float* __restrict__ h_raw)
{
    const int bh   = blockIdx.x >> 3;          // 0..255
    const int rg   = blockIdx.x & 7;           // row group of 64
    const int b    = bh >> 2, h = bh & 3;
    const int wave = threadIdx.x >> 5;
    const int lane = threadIdx.x & 31;

    const float it    = it_g[bh];
    const float ft    = ft_g[bh];
    const float inv_d = 1.0f / denom[bh];

    const size_t cbase = (size_t)bh * HD * HD;
    const int    vbase = b * HIDDEN + h * HD;
    const float4* q4 = (const float4*)(qvec + vbase);
    const float4* k4 = (const float4*)(kvec + vbase);

    for (int rr = 0; rr < 8; ++rr) {
        const int row = rg * 64 + wave * 8 + rr;
        const float vi = vvec[vbase + row] * it;
        const float4* cp = (const float4*)(C_prev + cbase + (size_t)row * HD);
        float4*       co = (float4*)(Ct_out + cbase + (size_t)row * HD);

        if (row < HD - 1) {  // uniform per wave: software prefetch next row
            const float* pf = C_prev + cbase + (size_t)(row + 1) * HD + lane * 16;
            __builtin_prefetch(pf, 0, 3);
        }

        float hacc = 0.f;
        #pragma unroll
        for (int t = 0; t < 4; ++t) {
            const int j4 = lane + t * 32;      // 128 float4 per row
            const float4 c  = cp[j4];
            const float4 kk = k4[j4];
            const float4 qq = q4[j4];
            float4 ct;
            ct.x = ft * c.x + vi * kk.x;
            ct.y = ft * c.y + vi * kk.y;
            ct.z = ft * c.z + vi * kk.z;
            ct.w = ft * c.w + vi * kk.w;
            co[j4] = ct;
            hacc += ct.x * qq.x + ct.y * qq.y + ct.z * qq.z + ct.w * qq.w;
        }
        // wave32 butterfly reduction
        for (int off = 16; off > 0; off >>= 1)
            hacc += __shfl_xor(hacc, off, 32);
        if (lane == 0) h_raw[vbase + row] = hacc * inv_d;   // h_tilde
    }
}

// ---------------- 7. output gate + per-head groupnorm + swish gating -------
__global__ __launch_bounds__(256) void gate_norm_kernel(
    const float* __restrict__ h_raw,           // h_tilde [64,2048]
    const float* __restrict__ o_t,             // o_tilda [64,2048]
    const float* __restrict__ skip,            // [64,2048]
    const float* __restrict__ gn_w, const float* __restrict__ gn_b,
    const float* __restrict__ xup,             // x_g = cols [2048,4096)
    float* __restrict__ g_out)                 // s * swish(x_g)
{
    __shared__ float red[256];
    const int bh = blockIdx.x;                 // 0..255
    const int b  = bh >> 2, h = bh & 3;
    const int base = b * HIDDEN + h * HD;

    float ht[2];
    float ssum = 0.f;
    #pragma unroll
    for (int t = 0; t < 2; ++t) {
        const int j = threadIdx.x + t * 256;
        const float ov = o_t[base + j];
        const float sig = 1.0f / (1.0f + __expf(-ov));
        ht[t] = sig * h_raw[base + j];
        ssum += ht[t];
    }
    const float mu = block_reduce_sum(ssum, red) * (1.0f / HD);

    float vsum = 0.f;
    #pragma unroll
    for (int t = 0; t < 2; ++t) { float d = ht[t] - mu; vsum += d * d; }
    const float rstd = rsqrtf(block_reduce_sum(vsum, red) * (1.0f / HD) + EPSV);

    #pragma unroll
    for (int t = 0; t < 2; ++t) {
        const int j = threadIdx.x + t * 256;
        const int hidx = h * HD + j;
        const float hn = (ht[t] - mu) * rstd * gn_w[hidx] + gn_b[hidx];
        const float s  = hn + skip[base + j];
        const float xg = xup[(size_t)b * UP_ROWS + PROJ + hidx];
        const float sw = xg / (1.0f + __expf(-xg));        // swish
        g_out[base + j] = s * sw;
    }
}

// ---------------------------------------------------------------------------
extern "C" void kernel_launch(void* const* d_in, const int* in_sizes, int n_in,
                              void* d_out, int out_size, void* d_ws, size_t ws_size,
                              hipStream_t stream) {
    (void)in_sizes; (void)n_in; (void)out_size; (void)ws_size;

    const float* x      = (const float*)d_in[0];
    const float* C_prev = (const float*)d_in[1];
    const float* n_prev = (const float*)d_in[2];
    const float* m_prev = (const float*)d_in[3];
    const float* ln_w   = (const float*)d_in[4];
    const float* ln_b   = (const float*)d_in[5];
    const float* up_w   = (const float*)d_in[6];
    const float* up_b   = (const float*)d_in[7];
    const float* wq     = (const float*)d_in[8];
    const float* bq     = (const float*)d_in[9];
    const float* wk     = (const float*)d_in[10];
    const float* bk     = (const float*)d_in[11];
    const float* wv     = (const float*)d_in[12];
    const float* bv     = (const float*)d_in[13];
    const float* wi     = (const float*)d_in[14];
    const float* bi     = (const float*)d_in[15];
    const float* wf     = (const float*)d_in[16];
    const float* bf     = (const float*)d_in[17];
    const float* wo     = (const float*)d_in[18];
    const float* bo     = (const float*)d_in[19];
    const float* skip_w = (const float*)d_in[20];
    const float* down_w = (const float*)d_in[21];
    const float* down_b = (const float*)d_in[22];
    const float* gn_w   = (const float*)d_in[23];
    const float* gn_b   = (const float*)d_in[24];

    // ---- workspace layout (floats) ----
    float* ws = (float*)d_ws;
    float* xn    = ws;                         // 64*1024
    float* x_up  = xn    + (size_t)BATCH * IN_DIM;     // 64*4096
    float* qv    = x_up  + (size_t)BATCH * UP_ROWS;    // 64*2048
    float* kv    = qv    + (size_t)BATCH * HIDDEN;
    float* vv    = kv    + (size_t)BATCH * HIDDEN;
    float* ot    = vv    + (size_t)BATCH * HIDDEN;
    float* skipv = ot    + (size_t)BATCH * HIDDEN;
    float* h_raw = skipv + (size_t)BATCH * HIDDEN;
    float* gbuf  = h_raw + (size_t)BATCH * HIDDEN;
    float* i_t   = gbuf  + (size_t)BATCH * HIDDEN;     // 256
    float* f_t   = i_t   + 256;
    float* it_g  = f_t   + 256;
    float* ft_g  = it_g  + 256;
    float* den   = ft_g  + 256;

    // ---- output layout (tuple concatenated): y, Ct, nt, mt ----
    float* y_out  = (float*)d_out;                               // 64*1024
    float* Ct_out = y_out  + (size_t)BATCH * IN_DIM;             // 64*4*512*512
    float* nt_out = Ct_out + (size_t)BATCH * HEADS * HD * HD;    // 64*4*512
    float* mt_out = nt_out + (size_t)BATCH * HEADS * HD;         // 64*4

    const float DIM_NORM = 1.0f / sqrtf((float)HIDDEN);

    // 1. layernorm
    layernorm_kernel<<<BATCH, 256, 0, stream>>>(x, ln_w, ln_b, xn);

    // 2. up-projection:  x_up[64,4096] = xn @ up_w^T + up_b
    gemm_wmma_kernel<<<UP_ROWS / 32, 256, 0, stream>>>(
        xn, IN_DIM, up_w, up_b, nullptr, x_up, IN_DIM, UP_ROWS, 1.0f);

    // 3. q/k/v/o/skip projections from x_qkv = x_up[:, :2048] (lda=4096)
    gemm_wmma_kernel<<<HIDDEN / 32, 256, 0, stream>>>(
        x_up, UP_ROWS, wq, bq, nullptr, qv, PROJ, HIDDEN, 1.0f);
    gemm_wmma_kernel<<<HIDDEN / 32, 256, 0, stream>>>(
        x_up, UP_ROWS, wk, bk, nullptr, kv, PROJ, HIDDEN, DIM_NORM);
    gemm_wmma_kernel<<<HIDDEN / 32, 256, 0, stream>>>(
        x_up, UP_ROWS, wv, bv, nullptr, vv, PROJ, HIDDEN, 1.0f);
    gemm_wmma_kernel<<<HIDDEN / 32, 256, 0, stream>>>(
        x_up, UP_ROWS, wo, bo, nullptr, ot, PROJ, HIDDEN, 1.0f);
    gemm_wmma_kernel<<<HIDDEN / 32, 256, 0, stream>>>(
        x_up, UP_ROWS, skip_w, nullptr, nullptr, skipv, PROJ, HIDDEN, 1.0f);

    // 4. i/f gate GEMVs + gate nonlinearity
    if_gemv_kernel<<<512, 256, 0, stream>>>(x_up, wi, bi, wf, bf, i_t, f_t);
    gates_kernel<<<1, 256, 0, stream>>>(i_t, f_t, m_prev, mt_out, it_g, ft_g);

    // 5. n state + denominator
    n_kernel<<<BATCH * HEADS, 256, 0, stream>>>(
        n_prev, kv, qv, it_g, ft_g, nt_out, den);

    // 6. C state update fused with h_tilde  (the 537MB streaming pass)
    c_update_kernel<<<BATCH * HEADS * 8, 256, 0, stream>>>(
        C_prev, kv, qv, vv, it_g, ft_g, den, Ct_out, h_raw);

    // 7. output gate + groupnorm + skip + swish gate
    gate_norm_kernel<<<BATCH * HEADS, 256, 0, stream>>>(
        h_raw, ot, skipv, gn_w, gn_b, x_up, gbuf);

    // 8. down-projection with bias + residual x:  y = g @ down_w^T + down_b + x
    gemm_wmma_kernel<<<IN_DIM / 32, 256, 0, stream>>>(
        gbuf, HIDDEN, down_w, down_b, x, y_out, HIDDEN, IN_DIM, 1.0f);
}